// Model_34119220199664
// MI455X (gfx1250) — compile-verified
//
#include <hip/hip_runtime.h>
#include <hip/hip_bf16.h>

// ---------------------------------------------------------------------------
// MoE layer for MI455X (gfx1250, wave32).  fp32 end-to-end: the workload is
// HBM-bound (~245MB weights @ 23.3TB/s ~= 10.5us floor; ~465 GFLOP of matrix
// work is far below the fp32 WMMA ceiling), so V_WMMA_F32_16X16X4_F32 keeps
// reference numerics at no cost.  LDS is double buffered and filled with
// gfx1250 async global->LDS copies (ASYNCcnt) so the next K-chunk's HBM fill
// overlaps the current chunk's WMMAs.
// ---------------------------------------------------------------------------

typedef __attribute__((ext_vector_type(2))) float v2f;
typedef __attribute__((ext_vector_type(8))) float v8f;
// exact pointee type the async-LDS builtin expects (from hipcc diagnostic):
typedef int b128_t __attribute__((vector_size(4 * sizeof(int))));

#define D_MODEL 1024
#define NUM_EXPERTS 8
#define INTER 2048
#define T_TOKENS 4096

#define TM 32          // block tile M (tokens)
#define TN 64          // block tile N (output features)
#define KC 32          // K chunk staged in LDS
#define LDA 36         // padded LDS row stride (floats): conflict-free, 16B-aligned rows

#if defined(__gfx1250__) &&                                                  \
    __has_builtin(__builtin_amdgcn_global_load_async_to_lds_b128) &&          \
    __has_builtin(__builtin_amdgcn_s_wait_asynccnt)
#define USE_ASYNC_LDS 1
#else
#define USE_ASYNC_LDS 0
#endif

// 16-byte global -> LDS copy.  Async path lands data in LDS without touching
// VGPRs and is tracked by ASYNCcnt; fallback is a plain load+ds_store.
__device__ __forceinline__ void cp16(const float* __restrict__ g, float* l) {
#if USE_ASYNC_LDS
    __builtin_amdgcn_global_load_async_to_lds_b128(
        (__attribute__((address_space(1))) b128_t*)(size_t)g,
        (__attribute__((address_space(3))) b128_t*)(size_t)(unsigned)(size_t)l,
        0, 0);
#else
    *(float4*)l = *(const float4*)g;
#endif
}

__device__ __forceinline__ void wait_async_all() {
#if USE_ASYNC_LDS
    __builtin_amdgcn_s_wait_asynccnt(0);
#endif
}

// ---------------------------------------------------------------------------
// Router: one wave per token.  logits = h @ router_w.T (written as gate_vals),
// softmax + top-2 -> dense combine[T,8], plus shared-expert sigmoid gate.
// ---------------------------------------------------------------------------
__global__ __launch_bounds__(256)
void moe_router_kernel(const float* __restrict__ h,
                       const float* __restrict__ router_w,
                       const float* __restrict__ sig_gate_w,
                       float* __restrict__ gate_out,   // [T,8]
                       float* __restrict__ combine,    // [T,8]
                       float* __restrict__ sig)        // [T]
{
    const int lane = threadIdx.x & 31;
    const int wv   = threadIdx.x >> 5;
    const int t    = blockIdx.x * 8 + wv;
    if (t >= T_TOKENS) return;

    const float* __restrict__ hr = h + (size_t)t * D_MODEL;
    float hl[D_MODEL / 32];
#pragma unroll
    for (int j = 0; j < D_MODEL / 32; ++j) hl[j] = hr[lane + j * 32];

    float logits[NUM_EXPERTS + 1];
#pragma unroll
    for (int e = 0; e < NUM_EXPERTS + 1; ++e) {
        const float* __restrict__ wr =
            (e < NUM_EXPERTS) ? (router_w + (size_t)e * D_MODEL) : sig_gate_w;
        float acc = 0.0f;
#pragma unroll
        for (int j = 0; j < D_MODEL / 32; ++j) acc += hl[j] * wr[lane + j * 32];
#pragma unroll
        for (int off = 16; off >= 1; off >>= 1) acc += __shfl_xor(acc, off, 32);
        logits[e] = acc;
    }

    float mx = logits[0];
#pragma unroll
    for (int e = 1; e < NUM_EXPERTS; ++e) mx = fmaxf(mx, logits[e]);
    float w[NUM_EXPERTS];
    float sum = 0.0f;
#pragma unroll
    for (int e = 0; e < NUM_EXPERTS; ++e) { w[e] = __expf(logits[e] - mx); sum += w[e]; }
    const float inv = 1.0f / sum;
#pragma unroll
    for (int e = 0; e < NUM_EXPERTS; ++e) w[e] *= inv;

    int i1 = 0;
#pragma unroll
    for (int e = 1; e < NUM_EXPERTS; ++e) if (w[e] > w[i1]) i1 = e;
    int i2 = (i1 == 0) ? 1 : 0;
#pragma unroll
    for (int e = 0; e < NUM_EXPERTS; ++e)
        if (e != i1 && w[e] > w[i2]) i2 = e;

    if (lane < NUM_EXPERTS) {
        gate_out[(size_t)t * NUM_EXPERTS + lane] = logits[lane];
        float c = (lane == i1) ? w[i1] : ((lane == i2) ? w[i2] : 0.0f);
        combine[(size_t)t * NUM_EXPERTS + lane] = c;
    }
    if (lane == 0) {
        sig[t] = 1.0f / (1.0f + __expf(-logits[NUM_EXPERTS]));
    }
}

// ---------------------------------------------------------------------------
// Fused gate/up GEMM:  tmp[t,n] = silu(h @ Wg^T) * (h @ Wu^T)
// Block = 256 threads = 8 waves; block tile 32(M) x 64(N); each wave owns a
// 16x16 tile of both g and u accumulators.  K staged through double-buffered
// LDS in chunks of 32: async fill of chunk k+1 overlaps WMMAs on chunk k.
// ---------------------------------------------------------------------------
__global__ __launch_bounds__(256)
void moe_gemm_gateup(const float* __restrict__ A,
                     const float* __restrict__ Bg,
                     const float* __restrict__ Bu,
                     float* __restrict__ Ctmp,   // [M x Ntot]
                     int Ntot, int K)
{
    __shared__ float As[2][TM * LDA];
    __shared__ float Bgs[2][TN * LDA];
    __shared__ float Bus[2][TN * LDA];

    const int m0 = blockIdx.x * TM;
    const int n0 = blockIdx.y * TN;
    const int tid  = threadIdx.x;
    const int lane = tid & 31;
    const int wv   = tid >> 5;
    const int msub = (wv >> 2) << 4;        // 0 / 16
    const int nsub = (wv & 3) << 4;         // 0 / 16 / 32 / 48
    const int lrow = lane & 15;
    const int khalf = (lane >> 4) << 1;     // 0 / 2

    const int fr = tid >> 3;                // 0..31
    const int fc = (tid & 7) << 2;          // 0,4,...,28

    v8f cg = {};
    v8f cu = {};

    auto load_chunk = [&](int kc, int buf) {
        const int k0 = kc * KC;
        cp16(&A[(size_t)(m0 + fr) * K + k0 + fc],        &As[buf][fr * LDA + fc]);
        cp16(&Bg[(size_t)(n0 + fr) * K + k0 + fc],       &Bgs[buf][fr * LDA + fc]);
        cp16(&Bg[(size_t)(n0 + fr + 32) * K + k0 + fc],  &Bgs[buf][(fr + 32) * LDA + fc]);
        cp16(&Bu[(size_t)(n0 + fr) * K + k0 + fc],       &Bus[buf][fr * LDA + fc]);
        cp16(&Bu[(size_t)(n0 + fr + 32) * K + k0 + fc],  &Bus[buf][(fr + 32) * LDA + fc]);
    };

    load_chunk(0, 0);
    wait_async_all();
    __syncthreads();

    const int nk = K / KC;
    for (int kc = 0; kc < nk; ++kc) {
        const int cur = kc & 1;
        if (kc + 1 < nk) load_chunk(kc + 1, cur ^ 1);   // overlaps with WMMAs

        const float* as  = &As[cur][(msub + lrow) * LDA + khalf];
        const float* bgs = &Bgs[cur][(nsub + lrow) * LDA + khalf];
        const float* bus = &Bus[cur][(nsub + lrow) * LDA + khalf];
#pragma unroll
        for (int kk = 0; kk < KC; kk += 4) {
            v2f a  = *(const v2f*)(as + kk);
            v2f bg = *(const v2f*)(bgs + kk);
            v2f bu = *(const v2f*)(bus + kk);
            cg = __builtin_amdgcn_wmma_f32_16x16x4_f32(
                     false, a, false, bg, (short)0, cg, false, false);
            cu = __builtin_amdgcn_wmma_f32_16x16x4_f32(
                     false, a, false, bu, (short)0, cu, false, false);
        }
        wait_async_all();
        __syncthreads();
    }

    // C layout: VGPR r -> row m+r (lanes 0-15) / m+r+8 (lanes 16-31)
    const int row_base = m0 + msub + ((lane >> 4) << 3);
    const int col = n0 + nsub + lrow;
#pragma unroll
    for (int r = 0; r < 8; ++r) {
        const float g = cg[r];
        const float u = cu[r];
        const float s = g / (1.0f + __expf(-g));   // silu
        Ctmp[(size_t)(row_base + r) * Ntot + col] = s * u;
    }
}

// ---------------------------------------------------------------------------
// Down-projection GEMM with per-row scale:  out[t,d] (+)= scale[t] * (A @ B^T)
// ---------------------------------------------------------------------------
template <bool ACCUM>
__global__ __launch_bounds__(256)
void moe_gemm_down(const float* __restrict__ A,
                   const float* __restrict__ B,
                   float* __restrict__ Out,     // [M x Ntot]
                   const float* __restrict__ Scale,
                   int sstride, int Ntot, int K)
{
    __shared__ float As[2][TM * LDA];
    __shared__ float Bs[2][TN * LDA];

    const int m0 = blockIdx.x * TM;
    const int n0 = blockIdx.y * TN;
    const int tid  = threadIdx.x;
    const int lane = tid & 31;
    const int wv   = tid >> 5;
    const int msub = (wv >> 2) << 4;
    const int nsub = (wv & 3) << 4;
    const int lrow = lane & 15;
    const int khalf = (lane >> 4) << 1;

    const int fr = tid >> 3;
    const int fc = (tid & 7) << 2;

    v8f c = {};

    auto load_chunk = [&](int kc, int buf) {
        const int k0 = kc * KC;
        cp16(&A[(size_t)(m0 + fr) * K + k0 + fc],       &As[buf][fr * LDA + fc]);
        cp16(&B[(size_t)(n0 + fr) * K + k0 + fc],       &Bs[buf][fr * LDA + fc]);
        cp16(&B[(size_t)(n0 + fr + 32) * K + k0 + fc],  &Bs[buf][(fr + 32) * LDA + fc]);
    };

    load_chunk(0, 0);
    wait_async_all();
    __syncthreads();

    const int nk = K / KC;
    for (int kc = 0; kc < nk; ++kc) {
        const int cur = kc & 1;
        if (kc + 1 < nk) load_chunk(kc + 1, cur ^ 1);

        const float* as = &As[cur][(msub + lrow) * LDA + khalf];
        const float* bs = &Bs[cur][(nsub + lrow) * LDA + khalf];
#pragma unroll
        for (int kk = 0; kk < KC; kk += 4) {
            v2f a = *(const v2f*)(as + kk);
            v2f b = *(const v2f*)(bs + kk);
            c = __builtin_amdgcn_wmma_f32_16x16x4_f32(
                    false, a, false, b, (short)0, c, false, false);
        }
        wait_async_all();
        __syncthreads();
    }

    const int row_base = m0 + msub + ((lane >> 4) << 3);
    const int col = n0 + nsub + lrow;
#pragma unroll
    for (int r = 0; r < 8; ++r) {
        const int row = row_base + r;
        const float sc = Scale[(size_t)row * sstride];
        const size_t idx = (size_t)row * Ntot + col;
        if (ACCUM)
            Out[idx] += sc * c[r];
        else
            Out[idx] = sc * c[r];
    }
}

// ---------------------------------------------------------------------------
// Launch: router -> shared expert (initializes out) -> 8 experts accumulate.
// Stream ordering serializes the read-modify-write on `out` and reuse of tmp.
// ---------------------------------------------------------------------------
extern "C" void kernel_launch(void* const* d_in, const int* in_sizes, int n_in,
                              void* d_out, int out_size, void* d_ws, size_t ws_size,
                              hipStream_t stream) {
    const float* h    = (const float*)d_in[0];   // [2,2048,1024] -> [T, D]
    const float* rw   = (const float*)d_in[1];   // [8, 1024]
    const float* eg   = (const float*)d_in[2];   // [8, 2048, 1024]
    const float* eu   = (const float*)d_in[3];   // [8, 2048, 1024]
    const float* ed   = (const float*)d_in[4];   // [8, 1024, 2048]
    const float* sg   = (const float*)d_in[5];   // [2048, 1024]
    const float* su   = (const float*)d_in[6];   // [2048, 1024]
    const float* sd   = (const float*)d_in[7];   // [1024, 2048]
    const float* ssg  = (const float*)d_in[8];   // [1, 1024]
    (void)in_sizes; (void)n_in; (void)out_size; (void)ws_size;

    float* out      = (float*)d_out;                           // [T, D]
    float* gate_out = out + (size_t)T_TOKENS * D_MODEL;        // [T, 8]

    float* ws      = (float*)d_ws;
    float* combine = ws;                                       // [T, 8]
    float* sig     = ws + (size_t)T_TOKENS * NUM_EXPERTS;      // [T]
    float* tmp     = ws + 40960;                               // [T, INTER]

    moe_router_kernel<<<T_TOKENS / 8, 256, 0, stream>>>(h, rw, ssg, gate_out,
                                                        combine, sig);

    const dim3 blk(256);
    const dim3 g1(T_TOKENS / TM, INTER / TN);
    const dim3 g2(T_TOKENS / TM, D_MODEL / TN);

    // shared expert first: writes out (no accumulate), scaled by sigmoid gate
    moe_gemm_gateup<<<g1, blk, 0, stream>>>(h, sg, su, tmp, INTER, D_MODEL);
    moe_gemm_down<false><<<g2, blk, 0, stream>>>(tmp, sd, out, sig, 1,
                                                 D_MODEL, INTER);

    // routed experts accumulate with their combine weights
    for (int e = 0; e < NUM_EXPERTS; ++e) {
        const float* Bg = eg + (size_t)e * INTER * D_MODEL;
        const float* Bu = eu + (size_t)e * INTER * D_MODEL;
        const float* Bd = ed + (size_t)e * D_MODEL * INTER;
        moe_gemm_gateup<<<g1, blk, 0, stream>>>(h, Bg, Bu, tmp, INTER, D_MODEL);
        moe_gemm_down<true><<<g2, blk, 0, stream>>>(tmp, Bd, out,
                                                    combine + e, NUM_EXPERTS,
                                                    D_MODEL, INTER);
    }
}